// HashEncoder_58007828300174
// MI455X (gfx1250) — compile-verified
//
#include <hip/hip_runtime.h>
#include <stdint.h>

// Hash-grid constants (compile-time; match reference)
//   D=3, L=16, C=2, H=16, T=2^19, BOUND=1
// Level table offsets: levels 0..2 dense ((res+1)^3), levels 3..15 hashed (2^19 each)
static constexpr uint32_t OFFS[16] = {
    0u,       4913u,    40850u,   315475u,
    839763u,  1364051u, 1888339u, 2412627u,
    2936915u, 3461203u, 3985491u, 4509779u,
    5034067u, 5558355u, 6082643u, 6606931u};

#define PRIME1 2654435761u
#define PRIME2 805459861u
#define HASH_MASK 0x7FFFFu  // T - 1 = 2^19 - 1

typedef __attribute__((ext_vector_type(4))) float v4f;
typedef __attribute__((ext_vector_type(2))) float v2f;

__global__ __launch_bounds__(256) void HashEncoder_gfx1250_kernel(
    const float* __restrict__ inputs,   // [n, 3]
    const float* __restrict__ emb,      // [7131219, 2]
    float* __restrict__ out,            // [n, 32]
    int n)
{
    const int i = blockIdx.x * blockDim.x + threadIdx.x;
    if (i >= n) return;

    // CDNA5 path: speculative L2 prefetch of input rows ~256 waves (32 blocks) ahead
    // (emits global_prefetch_b8; translation failures silently dropped)
    {
        const int pf = i + 8192;
        if (pf < n) __builtin_prefetch(inputs + 3ull * pf, 0, 0);
    }

    // Streamed-once input: non-temporal loads (TH=NT) so the 12 MB input
    // stream never competes with the L2-resident embedding table.
    const float vx = __builtin_nontemporal_load(inputs + 3ull * i + 0);
    const float vy = __builtin_nontemporal_load(inputs + 3ull * i + 1);
    const float vz = __builtin_nontemporal_load(inputs + 3ull * i + 2);

    // normalize to [0,1]: clip((v + 1) / 2, 0, 1)
    const float x = fminf(fmaxf((vx + 1.0f) * 0.5f, 0.0f), 1.0f);
    const float y = fminf(fmaxf((vy + 1.0f) * 0.5f, 0.0f), 1.0f);
    const float z = fminf(fmaxf((vz + 1.0f) * 0.5f, 0.0f), 1.0f);

    const v2f* __restrict__ e2 = reinterpret_cast<const v2f*>(emb);

    v2f acc[16];

#pragma unroll
    for (int l = 0; l < 16; ++l) {
        const uint32_t resi = (16u << l);
        const float    res  = (float)resi;
        const uint32_t R1   = resi + 1u;          // dense stride base

        // Per-level table base is a compile-time SGPR constant: gathers become
        // global_load_b64 vdst, v_idx, s[base] scale_offset (no per-gather VALU add).
        const v2f* __restrict__ tab = e2 + OFFS[l];

        const float px = x * res, py = y * res, pz = z * res;
        // pg = clip(floor(pos), 0, res-1); pos >= 0 so only upper clip needed
        const float gx = fminf(floorf(px), res - 1.0f);
        const float gy = fminf(floorf(py), res - 1.0f);
        const float gz = fminf(floorf(pz), res - 1.0f);
        const float fx = px - gx, fy = py - gy, fz = pz - gz;
        const uint32_t ix = (uint32_t)gx;
        const uint32_t iy = (uint32_t)gy;
        const uint32_t iz = (uint32_t)gz;

        // corner k: bit2 -> dim0(x), bit1 -> dim1(y), bit0 -> dim2(z)
        uint32_t idx[8];
        if (l < 3) {
            // dense: idx = x + y*(res+1) + z*(res+1)^2   (always < table size)
            const uint32_t R2   = R1 * R1;
            const uint32_t base = ix + iy * R1 + iz * R2;
            idx[0] = base;
            idx[1] = base + R2;
            idx[2] = base + R1;
            idx[3] = base + R1 + R2;
            idx[4] = base + 1u;
            idx[5] = base + 1u + R2;
            idx[6] = base + 1u + R1;
            idx[7] = base + 1u + R1 + R2;
        } else {
            // spatial hash: (x*1) ^ (y*P1) ^ (z*P2), mod 2^19 == & mask
            const uint32_t hx0 = ix,                 hx1 = ix + 1u;
            const uint32_t hy0 = iy * PRIME1,        hy1 = (iy + 1u) * PRIME1;
            const uint32_t hz0 = iz * PRIME2,        hz1 = (iz + 1u) * PRIME2;
            idx[0] = (hx0 ^ hy0 ^ hz0) & HASH_MASK;
            idx[1] = (hx0 ^ hy0 ^ hz1) & HASH_MASK;
            idx[2] = (hx0 ^ hy1 ^ hz0) & HASH_MASK;
            idx[3] = (hx0 ^ hy1 ^ hz1) & HASH_MASK;
            idx[4] = (hx1 ^ hy0 ^ hz0) & HASH_MASK;
            idx[5] = (hx1 ^ hy0 ^ hz1) & HASH_MASK;
            idx[6] = (hx1 ^ hy1 ^ hz0) & HASH_MASK;
            idx[7] = (hx1 ^ hy1 ^ hz1) & HASH_MASK;
        }

        // 8 x global_load_b64 gathers (L2-resident table, default RT policy)
        v2f f[8];
#pragma unroll
        for (int k = 0; k < 8; ++k)
            f[k] = tab[idx[k]];

        // trilinear weights: w[k] = prod_d (bit ? frac : 1-frac), (x*y)*z order
        const float wx0 = 1.0f - fx, wx1 = fx;
        const float wy0 = 1.0f - fy, wy1 = fy;
        const float wz0 = 1.0f - fz, wz1 = fz;
        const float w00 = wx0 * wy0, w01 = wx0 * wy1;
        const float w10 = wx1 * wy0, w11 = wx1 * wy1;
        const float wk[8] = {w00 * wz0, w00 * wz1, w01 * wz0, w01 * wz1,
                             w10 * wz0, w10 * wz1, w11 * wz0, w11 * wz1};

        // Packed interpolation: both channels per corner in one v_pk_fma_f32
        // (gathered float2 already lives in an aligned VGPR pair).
        v2f a = {0.0f, 0.0f};
#pragma unroll
        for (int k = 0; k < 8; ++k) {
            const v2f wv = {wk[k], wk[k]};
            a = __builtin_elementwise_fma(wv, f[k], a);
        }
        acc[l] = a;
    }

    // 32 contiguous floats per point -> 8 x global_store_b128 with TH=NT:
    // write-once output must not evict the embedding table from L2.
    v4f* __restrict__ o = reinterpret_cast<v4f*>(out + 32ull * i);
#pragma unroll
    for (int q = 0; q < 8; ++q) {
        v4f r = {acc[2 * q].x, acc[2 * q].y, acc[2 * q + 1].x, acc[2 * q + 1].y};
        __builtin_nontemporal_store(r, o + q);
    }
}

extern "C" void kernel_launch(void* const* d_in, const int* in_sizes, int n_in,
                              void* d_out, int out_size, void* d_ws, size_t ws_size,
                              hipStream_t stream) {
    (void)n_in; (void)d_ws; (void)ws_size; (void)out_size;
    const float* inputs = (const float*)d_in[0];   // [B,3] f32
    const float* emb    = (const float*)d_in[1];   // [7131219,2] f32
    float* out          = (float*)d_out;           // [B,32] f32

    const int n = in_sizes[0] / 3;
    const int block = 256;
    const int grid  = (n + block - 1) / block;
    HashEncoder_gfx1250_kernel<<<grid, block, 0, stream>>>(inputs, emb, out, n);
}